// DAGCN_reduce_7705171329817
// MI455X (gfx1250) — compile-verified
//
#include <hip/hip_runtime.h>
#include <hip/hip_bf16.h>
#include <cstddef>

// ---------------------------------------------------------------------------
// DAGCN adaptive graph conv for MI455X (gfx1250, wave32, WMMA f32 16x16x4).
//
// T=12, N=1024, D=10, K=3, C=O=32, B=16.
// Algebra: sup[0]=I, sup[2]=2*diag(S)⊙S - I  =>  only y = S@x is a real
// N x N matmul; the final per-node GEMM folds to K=64 with two effective
// 32x32 theta matrices:
//   thA = sum_d e_d (W0-W2)         (applied to x rows)
//   thB = sum_d e_d (W1 + 2*diag*W2) (applied to y rows)
//
// Scratch layout in d_ws (floats):
//   S    : [T][N][N]       = 12,582,912   (supports, softmax'd in place)
//   Y    : [T][N][B][C]    =  6,291,456   (y = S @ x)
//   diag : [T][N]          =     12,288
//   WA   : [T][C][O][12]   =    147,456   (W0-W2, d padded 10->12)
//   WB   : [T][C][O][24]   =    294,912   ({W1,W2} interleaved, d padded)
// total ~77.3 MB.
// ---------------------------------------------------------------------------

namespace {
constexpr int cT = 12;
constexpr int cN = 1024;
constexpr int cD = 10;
constexpr int cC = 32;
constexpr int cO = 32;
constexpr int cB = 16;
}

typedef __attribute__((ext_vector_type(2))) float v2f;
typedef __attribute__((ext_vector_type(8))) float v8f;

// f32 WMMA fragment conventions (wave32), per CDNA5 ISA 7.12.2:
//   A (16x4 MxK): VGPR v, lane l -> M = l%16, K = v + 2*(l>=16)
//   B (4x16 KxN): VGPR v, lane l -> N = l%16, K = v + 2*(l>=16)
//   C/D (16x16) : VGPR r, lane l -> M = r + 8*(l>=16), N = l%16

// ---------------------------------------------------------------------------
// Kernel 0: repack weights_pool into WA / WB (d-innermost, padded to 12).
//   WA[t][i][o][d]      = W[t,d,0,i,o] - W[t,d,2,i,o]          (d<10, else 0)
//   WB[t][i][o][2d+0/1] = W[t,d,1,i,o] / W[t,d,2,i,o]          (d<10, else 0)
// ---------------------------------------------------------------------------
__global__ void dagcn_repack_w(const float* __restrict__ W,
                               float* __restrict__ WA,
                               float* __restrict__ WB) {
    const int tid = blockIdx.x * blockDim.x + threadIdx.x;
    if (tid >= cT * cC * cO) return;
    const int t  = tid / (cC * cO);
    const int io = tid % (cC * cO);
    const int i  = io / cO;
    const int o  = io % cO;

    float* wa = WA + (size_t)tid * 12;
    float* wb = WB + (size_t)tid * 24;
    #pragma unroll
    for (int d = 0; d < 12; ++d) {
        float w0 = 0.0f, w1 = 0.0f, w2 = 0.0f;
        if (d < cD) {
            const size_t base =
                (((size_t)t * cD + d) * 3) * (cC * cO) + (size_t)i * cO + o;
            w0 = W[base];
            w1 = W[base + (size_t)cC * cO];
            w2 = W[base + (size_t)2 * cC * cO];
        }
        wa[d]         = w0 - w2;
        wb[2 * d]     = w1;
        wb[2 * d + 1] = w2;
    }
}

// ---------------------------------------------------------------------------
// Kernel 1: S[t][n][s] = relu( sum_d E[t][n][d] * E[t][s][d] )
// One wave computes a 16(n) x 64(s) strip; K=10 padded to 12 (3 WMMA steps).
// Tail step loads d=8,9 unconditionally (always in range) and masks the
// padding lanes with selects -> no exec divergence.
// ---------------------------------------------------------------------------
__global__ void dagcn_gram_relu(const float* __restrict__ E, float* __restrict__ S) {
    const int wavesPerBlock = blockDim.x >> 5;
    const int wid  = blockIdx.x * wavesPerBlock + (threadIdx.x >> 5);
    const int lane = threadIdx.x & 31;
    constexpr int stripsPerT = (cN / 16) * (cN / 64);   // 64 * 16 = 1024
    const int t = wid / stripsPerT;
    if (t >= cT) return;                                 // wave-uniform
    const int r  = wid % stripsPerT;
    const int n0 = (r / (cN / 64)) * 16;
    const int s0 = (r % (cN / 64)) * 64;

    const float* Et = E + (size_t)t * cN * cD;
    const int row   = lane & 15;
    const int khalf = (lane >> 4) << 1;                  // 0 or 2
    const bool lo   = (lane < 16);

    const float* arow = Et + (size_t)(n0 + row) * cD;

    v8f acc[4] = {};
    // steps 0,1: K rows 0..7, all in range
    #pragma unroll
    for (int step = 0; step < 2; ++step) {
        const int d0 = step * 4 + khalf;
        v2f a; a.x = arow[d0]; a.y = arow[d0 + 1];
        #pragma unroll
        for (int jt = 0; jt < 4; ++jt) {
            const int s = s0 + jt * 16 + row;
            v2f b;
            b.x = Et[(size_t)s * cD + d0];
            b.y = Et[(size_t)s * cD + d0 + 1];
            acc[jt] = __builtin_amdgcn_wmma_f32_16x16x4_f32(
                false, a, false, b, (short)0, acc[jt], false, false);
        }
    }
    // step 2: K rows 8..11; rows 10,11 are zero padding (high-half lanes)
    {
        const float ax = arow[8], ay = arow[9];
        v2f a; a.x = lo ? ax : 0.0f; a.y = lo ? ay : 0.0f;
        #pragma unroll
        for (int jt = 0; jt < 4; ++jt) {
            const int s = s0 + jt * 16 + row;
            const float bx = Et[(size_t)s * cD + 8];
            const float by = Et[(size_t)s * cD + 9];
            v2f b; b.x = lo ? bx : 0.0f; b.y = lo ? by : 0.0f;
            acc[jt] = __builtin_amdgcn_wmma_f32_16x16x4_f32(
                false, a, false, b, (short)0, acc[jt], false, false);
        }
    }
    float* St = S + (size_t)t * cN * cN;
    const int rbase = (lane >> 4) << 3;                  // 0 or 8
    #pragma unroll
    for (int jt = 0; jt < 4; ++jt) {
        const int col = s0 + jt * 16 + row;
        #pragma unroll
        for (int rr = 0; rr < 8; ++rr) {
            const float v = acc[jt][rr];
            St[(size_t)(n0 + rbase + rr) * cN + col] = v > 0.0f ? v : 0.0f;
        }
    }
}

// ---------------------------------------------------------------------------
// Kernel 2: in-place row softmax of S + diagonal extraction.
// One wave32 per row (1024 elems -> 32 per lane), shuffle reductions.
// ---------------------------------------------------------------------------
__global__ void dagcn_softmax_rows(float* __restrict__ S, float* __restrict__ diag) {
    const int wavesPerBlock = blockDim.x >> 5;
    const int wid  = blockIdx.x * wavesPerBlock + (threadIdx.x >> 5);
    const int lane = threadIdx.x & 31;
    if (wid >= cT * cN) return;                          // wave-uniform
    const int t = wid / cN;
    const int n = wid % cN;

    float* row = S + ((size_t)t * cN + n) * cN;
    float vals[cN / 32];
    float m = -3.0e38f;
    #pragma unroll
    for (int i = 0; i < cN / 32; ++i) {
        vals[i] = row[i * 32 + lane];
        m = fmaxf(m, vals[i]);
    }
    #pragma unroll
    for (int off = 16; off > 0; off >>= 1) m = fmaxf(m, __shfl_xor(m, off, 32));
    float sum = 0.0f;
    #pragma unroll
    for (int i = 0; i < cN / 32; ++i) {
        vals[i] = __expf(vals[i] - m);
        sum += vals[i];
    }
    #pragma unroll
    for (int off = 16; off > 0; off >>= 1) sum += __shfl_xor(sum, off, 32);
    const float inv = 1.0f / sum;
    #pragma unroll
    for (int i = 0; i < cN / 32; ++i) {
        const float v = vals[i] * inv;
        row[i * 32 + lane] = v;
        if (i * 32 + lane == n) diag[(size_t)t * cN + n] = v;
    }
}

// ---------------------------------------------------------------------------
// Kernel 3: Y[t][n][b][c] = sum_m S[t][n][m] * x[b][t][m][c]
// One wave computes a 16(n) x 64(j = b*32+c) strip; K loop over N in steps
// of 4; the A fragment (S row chunk) is reused across 4 B tiles.
// ---------------------------------------------------------------------------
__global__ void dagcn_aggregate(const float* __restrict__ S,
                                const float* __restrict__ x,
                                float* __restrict__ Y) {
    constexpr int BC = cB * cC;                          // 512
    const int wavesPerBlock = blockDim.x >> 5;
    const int wid  = blockIdx.x * wavesPerBlock + (threadIdx.x >> 5);
    const int lane = threadIdx.x & 31;
    constexpr int stripsPerT = (cN / 16) * (BC / 64);    // 64 * 8 = 512
    const int t = wid / stripsPerT;
    if (t >= cT) return;                                 // wave-uniform
    const int r  = wid % stripsPerT;
    const int n0 = (r / (BC / 64)) * 16;
    const int j0 = (r % (BC / 64)) * 64;

    const float* St  = S + (size_t)t * cN * cN;
    const int row    = lane & 15;
    const int khalf  = (lane >> 4) << 1;                 // 0 or 2

    int bcol[4], ccol[4];
    #pragma unroll
    for (int jt = 0; jt < 4; ++jt) {
        const int j = j0 + jt * 16 + row;
        bcol[jt] = j >> 5;                               // b
        ccol[jt] = j & 31;                               // c
    }

    v8f acc[4] = {};
    for (int m0 = 0; m0 < cN; m0 += 4) {
        const float* srow = St + (size_t)(n0 + row) * cN + m0 + khalf;
        v2f a; a.x = srow[0]; a.y = srow[1];
        #pragma unroll
        for (int jt = 0; jt < 4; ++jt) {
            const float* xp =
                x + (((size_t)bcol[jt] * cT + t) * cN + (m0 + khalf)) * cC + ccol[jt];
            v2f b; b.x = xp[0]; b.y = xp[cC];
            acc[jt] = __builtin_amdgcn_wmma_f32_16x16x4_f32(
                false, a, false, b, (short)0, acc[jt], false, false);
        }
    }
    const int rbase = (lane >> 4) << 3;
    #pragma unroll
    for (int jt = 0; jt < 4; ++jt) {
        #pragma unroll
        for (int rr = 0; rr < 8; ++rr) {
            const int nn = n0 + rbase + rr;
            Y[(((size_t)t * cN + nn) * cB + bcol[jt]) * cC + ccol[jt]] = acc[jt][rr];
        }
    }
}

// ---------------------------------------------------------------------------
// Kernel 4: per-(t,n) GEMM with on-the-fly theta from packed WA/WB, bias
// fused.  One wave per (t,n): M=16(b), N=32(o: 2 tiles), K=64 (x rows || y
// rows).  Theta B-fragments built from b128 loads of WA/WB (d innermost).
// ---------------------------------------------------------------------------
__global__ void dagcn_final(const float* __restrict__ x,
                            const float* __restrict__ E,
                            const float* __restrict__ WA,
                            const float* __restrict__ WB,
                            const float* __restrict__ bpool,
                            const float* __restrict__ Y,
                            const float* __restrict__ diag,
                            float* __restrict__ out) {
    const int wavesPerBlock = blockDim.x >> 5;
    const int wid  = blockIdx.x * wavesPerBlock + (threadIdx.x >> 5);
    const int lane = threadIdx.x & 31;
    if (wid >= cT * cN) return;                          // wave-uniform
    const int t = wid / cN;
    const int n = wid % cN;
    const int row   = lane & 15;
    const int khalf = (lane >> 4) << 1;

    float e[12], et2[12];
    const float twod = 2.0f * diag[(size_t)t * cN + n];
    #pragma unroll
    for (int d = 0; d < 12; ++d) {
        const float ed = (d < cD) ? E[((size_t)t * cN + n) * cD + d] : 0.0f;
        e[d]   = ed;
        et2[d] = twod * ed;
    }

    v8f acc[2] = {};
    #pragma unroll
    for (int ks = 0; ks < 16; ++ks) {
        const int kbase = ks * 4 + khalf;                // even; varies by lane half
        v2f a;
        if (ks < 8) {                                    // K rows from x
            const float* xp = x + (((size_t)row * cT + t) * cN + n) * cC + kbase;
            a.x = xp[0]; a.y = xp[1];
        } else {                                         // K rows from y
            const float* yp =
                Y + (((size_t)t * cN + n) * cB + row) * cC + (kbase - cC);
            a.x = yp[0]; a.y = yp[1];
        }
        #pragma unroll
        for (int ot = 0; ot < 2; ++ot) {
            const int o = ot * 16 + row;
            float th0 = 0.0f, th1 = 0.0f;
            if (ks < 8) {
                const int i = kbase;
                const float4* wa0 = reinterpret_cast<const float4*>(
                    WA + (((size_t)t * cC + i) * cO + o) * 12);
                const float4* wa1 = reinterpret_cast<const float4*>(
                    WA + (((size_t)t * cC + i + 1) * cO + o) * 12);
                #pragma unroll
                for (int q = 0; q < 3; ++q) {
                    const float4 w0 = wa0[q];
                    const float4 w1 = wa1[q];
                    th0 = fmaf(e[4*q+0], w0.x, fmaf(e[4*q+1], w0.y,
                          fmaf(e[4*q+2], w0.z, fmaf(e[4*q+3], w0.w, th0))));
                    th1 = fmaf(e[4*q+0], w1.x, fmaf(e[4*q+1], w1.y,
                          fmaf(e[4*q+2], w1.z, fmaf(e[4*q+3], w1.w, th1))));
                }
            } else {
                const int i = kbase - cC;
                const float4* wb0 = reinterpret_cast<const float4*>(
                    WB + (((size_t)t * cC + i) * cO + o) * 24);
                const float4* wb1 = reinterpret_cast<const float4*>(
                    WB + (((size_t)t * cC + i + 1) * cO + o) * 24);
                #pragma unroll
                for (int q = 0; q < 6; ++q) {            // pairs {w1,w2} x 2 d's
                    const float4 w0 = wb0[q];
                    const float4 w1 = wb1[q];
                    th0 = fmaf(e[2*q+0], w0.x, fmaf(et2[2*q+0], w0.y,
                          fmaf(e[2*q+1], w0.z, fmaf(et2[2*q+1], w0.w, th0))));
                    th1 = fmaf(e[2*q+0], w1.x, fmaf(et2[2*q+0], w1.y,
                          fmaf(e[2*q+1], w1.z, fmaf(et2[2*q+1], w1.w, th1))));
                }
            }
            v2f bf; bf.x = th0; bf.y = th1;
            acc[ot] = __builtin_amdgcn_wmma_f32_16x16x4_f32(
                false, a, false, bf, (short)0, acc[ot], false, false);
        }
    }
    const int rbase = (lane >> 4) << 3;
    #pragma unroll
    for (int ot = 0; ot < 2; ++ot) {
        const int o = ot * 16 + row;
        float bias = 0.0f;
        #pragma unroll
        for (int d = 0; d < cD; ++d)
            bias = fmaf(e[d], bpool[((size_t)t * cD + d) * cO + o], bias);
        #pragma unroll
        for (int rr = 0; rr < 8; ++rr) {
            const int b = rbase + rr;
            out[(((size_t)b * cT + t) * cN + n) * cO + o] = acc[ot][rr] + bias;
        }
    }
}

// ---------------------------------------------------------------------------
// Host launcher.  Inputs: x[B,T,N,C], dn_embeddings[T,N,D],
// weights_pool[T,D,3,C,O], bias_pool[T,D,O]; output [B,T,N,O], all fp32.
// ---------------------------------------------------------------------------
extern "C" void kernel_launch(void* const* d_in, const int* in_sizes, int n_in,
                              void* d_out, int out_size, void* d_ws, size_t ws_size,
                              hipStream_t stream) {
    const float* x  = (const float*)d_in[0];
    const float* E  = (const float*)d_in[1];
    const float* W  = (const float*)d_in[2];
    const float* bp = (const float*)d_in[3];
    float* out = (float*)d_out;

    float* S    = (float*)d_ws;                         // [T][N][N]
    float* Y    = S    + (size_t)cT * cN * cN;          // [T][N][B][C]
    float* diag = Y    + (size_t)cT * cN * cB * cC;     // [T][N]
    float* WA   = diag + (size_t)cT * cN;               // [T][C][O][12]
    float* WB   = WA   + (size_t)cT * cC * cO * 12;     // [T][C][O][24]
    // required ws ~ 77.3 MB
    (void)in_sizes; (void)n_in; (void)out_size; (void)ws_size;

    const int threads = 128;                            // 4 waves / block
    const int wavesPerBlock = threads / 32;

    {   // Kernel 0: weight repack (independent of kernels 1-3)
        const int tot = cT * cC * cO;
        dagcn_repack_w<<<(tot + threads - 1) / threads, threads, 0, stream>>>(W, WA, WB);
    }
    {   // Kernel 1: 12 * (1024/16) * (1024/64) = 12288 waves
        const int waves = cT * (cN / 16) * (cN / 64);
        dagcn_gram_relu<<<waves / wavesPerBlock, threads, 0, stream>>>(E, S);
    }
    {   // Kernel 2: one wave per row -> 12288 waves
        const int waves = cT * cN;
        dagcn_softmax_rows<<<waves / wavesPerBlock, threads, 0, stream>>>(S, diag);
    }
    {   // Kernel 3: 12 * (1024/16) * (512/64) = 6144 waves
        const int waves = cT * (cN / 16) * ((cB * cC) / 64);
        dagcn_aggregate<<<waves / wavesPerBlock, threads, 0, stream>>>(S, x, Y);
    }
    {   // Kernel 4: one wave per (t,n) -> 12288 waves
        const int waves = cT * cN;
        dagcn_final<<<waves / wavesPerBlock, threads, 0, stream>>>(
            x, E, WA, WB, bp, Y, diag, out);
    }
}